// TemporalSamplingUpSampler_51445118271815
// MI455X (gfx1250) — compile-verified
//
#include <hip/hip_runtime.h>
#include <hip/hip_bf16.h>
#include <stdint.h>

// CDNA5 / gfx1250. Store-bound broadcast-gather: out[c,t] = A[c,k(t)] * m(t).
// Uses the Tensor Data Mover (tensor_load_to_lds + s_wait_tensorcnt) to stage
// the A tile into LDS, then streams coalesced 128-bit stores.

typedef unsigned int u32x4 __attribute__((ext_vector_type(4)));
typedef int          i32x8 __attribute__((ext_vector_type(8)));
typedef int          i32x4 __attribute__((ext_vector_type(4)));

#define K_SEG 256   // K (segments)
#define C_ALL 256   // C (channels)
#define C_PB  128   // channels per block (blockIdx.y halves)
#define T_PB  1024  // t-slots per block (256 threads x 4)

// ---------------------------------------------------------------------------
// Kernel 1: planning (softmax -> Lp, r=rint, prefix sums, l_max, total)
// ---------------------------------------------------------------------------
__global__ void __launch_bounds__(256)
plan_kernel(const float* __restrict__ L, int T,
            int* __restrict__ cum, int* __restrict__ meta,
            float* __restrict__ LpOut) {
  __shared__ float red[K_SEG];
  __shared__ int   r_s[K_SEG];
  const int k = threadIdx.x;

  float lv = L[k];
  red[k] = lv; __syncthreads();
  for (int s = K_SEG / 2; s > 0; s >>= 1) {
    if (k < s) red[k] = fmaxf(red[k], red[k + s]);
    __syncthreads();
  }
  float mx = red[0];
  __syncthreads();

  float e = expf(lv - mx);
  red[k] = e; __syncthreads();
  for (int s = K_SEG / 2; s > 0; s >>= 1) {
    if (k < s) red[k] = red[k] + red[k + s];
    __syncthreads();
  }
  float sum = red[0];
  __syncthreads();

  float Lp = (float)T * (e / sum);            // T * softmax(L)
  int r = (int)rintf(Lp);                     // round-half-even == np.rint
  if (r < 0) r = 0;
  r_s[k] = r;

  red[k] = Lp; __syncthreads();               // max(Lp) for l_max
  for (int s = K_SEG / 2; s > 0; s >>= 1) {
    if (k < s) red[k] = fmaxf(red[k], red[k + s]);
    __syncthreads();
  }

  if (k == 0) {
    int c = 0;
    for (int i = 0; i < K_SEG; ++i) { cum[i] = c; c += r_s[i]; }
    cum[K_SEG] = c;
    meta[0] = (int)(red[0] + 0.5f);           // l_max = int(max(Lp)+0.5)
    meta[1] = c;                              // total
  }
  LpOut[k] = Lp;
}

// ---------------------------------------------------------------------------
// Kernel 2: main sampler. blockIdx.x -> 1024-wide t-tile, blockIdx.y -> c-half
// ---------------------------------------------------------------------------
__global__ void __launch_bounds__(256)
sample_kernel(const float* __restrict__ A,
              const int* __restrict__ cum,
              const int* __restrict__ meta,
              const float* __restrict__ Lp,
              float* __restrict__ out, int T) {
  __shared__ float A_lds[C_PB * K_SEG];       // 128 KB: rows c-local, cols k
  __shared__ int   cum_s[K_SEG + 1];
  __shared__ float Lp_s[K_SEG];
  __shared__ int   meta_s[2];

  const int  tid   = threadIdx.x;             // 0..255
  const int  cHalf = blockIdx.y;              // 0..1
  const long t0    = (long)blockIdx.x * T_PB;

  if (tid < 2)     meta_s[tid] = meta[tid];
  if (tid < K_SEG) { cum_s[tid] = cum[tid]; Lp_s[tid] = Lp[tid]; }
  if (tid == 0)    cum_s[K_SEG] = cum[K_SEG];

  // ---- TDM: DMA A rows [cHalf*128, cHalf*128+128) x 256 cols into LDS -----
  if (tid == 0) {
    unsigned long long gaddr = (unsigned long long)(uintptr_t)(const void*)A
                             + (unsigned long long)cHalf * (unsigned long long)(C_PB * K_SEG * 4);
    unsigned lds_off = (unsigned)(uintptr_t)(void*)&A_lds[0]; // low 32b = LDS byte offset

    u32x4 g0; i32x8 g1; i32x4 g2 = {0, 0, 0, 0}; i32x4 g3 = {0, 0, 0, 0};
    // D# group 0: count=1 (valid), lds_addr, global_addr[56:0], type=2
    g0[0] = 1u;
    g0[1] = lds_off;
    g0[2] = (unsigned)(gaddr & 0xFFFFFFFFull);
    g0[3] = (unsigned)((gaddr >> 32) & 0x1FFFFFFull) | (2u << 30);
    // D# group 1: data_size=2 (4B); tensor 256x256, tile 256x128, stride0=256
    g1[0] = 0x20000;              // [17:16]=data_size=2; wg_mask=0 (not in cluster)
    g1[1] = (K_SEG << 16);        // tensor_dim0[15:0] at bits[63:48]
    g1[2] = (C_ALL << 16);        // tensor_dim0 hi=0 | tensor_dim1[15:0] at [95:80]
    g1[3] = (K_SEG << 16);        // tensor_dim1 hi=0 | tile_dim0=256 at [127:112]
    g1[4] = C_PB;                 // tile_dim1=128; tile_dim2=0
    g1[5] = K_SEG;                // tensor_dim0_stride low32 = 256
    g1[6] = 0;
    g1[7] = 0;                    // tensor_dim1_stride unused for 2D tile
#if defined(__clang_major__) && __clang_major__ >= 23
    i32x8 g4 = {0, 0, 0, 0, 0, 0, 0, 0};
    __builtin_amdgcn_tensor_load_to_lds(g0, g1, g2, g3, g4, 0);
#else
    __builtin_amdgcn_tensor_load_to_lds(g0, g1, g2, g3, 0);
#endif
    __builtin_amdgcn_s_wait_tensorcnt(0);
  }
  __syncthreads();

  const int    l_max = meta_s[0];
  const int    total = meta_s[1];
  const double scale = (double)total / (double)T;   // matches numpy total/T
  const float  fl    = (float)l_max;

  const long tb = t0 + (long)tid * 4;
  if (tb >= (long)T) return;

  int   kk[4];
  float mm[4];
#pragma unroll
  for (int j = 0; j < 4; ++j) {
    long t = tb + j;
    if (t >= (long)T) t = (long)T - 1;
    // nearest-interp source: i = min(floor(t * total/T), total-1)   (fp64)
    long i64 = (long)floor((double)t * scale);
    if (i64 > (long)total - 1) i64 = (long)total - 1;
    if (i64 < 0) i64 = 0;
    const int i = (int)i64;
    // largest k with cum[k] <= i  (cum[0]==0 so invariant holds)
    int lo = 0, hi = K_SEG - 1;
    while (lo < hi) {
      int mid = (lo + hi + 1) >> 1;
      if (cum_s[mid] <= i) lo = mid; else hi = mid - 1;
    }
    const int   k   = lo;
    const int   l   = i - cum_s[k];
    const float Lpk = Lp_s[k];
    // grid_sample bilinear edge weight on width-100 constant input (fp32)
    const float x   = (2.0f * (float)l + 1.0f) / fl - 1.0f;
    const float s   = fl / Lpk;
    const float tt  = (fl - Lpk) / Lpk;
    const float xs  = s * x + tt;
    const float p   = ((xs + 1.0f) * 100.0f - 1.0f) * 0.5f;
    const float p0  = floorf(p);
    const float w1  = p - p0;
    const float in0 = (p0 >= 0.0f  && p0 <= 99.0f) ? 1.0f : 0.0f;
    const float in1 = (p0 >= -1.0f && p0 <= 98.0f) ? 1.0f : 0.0f;
    kk[j] = k;
    mm[j] = (1.0f - w1) * in0 + w1 * in1;
  }

  float* obase = out + (long)cHalf * C_PB * (long)T + tb;
  const bool full = (tb + 3) < (long)T;
#pragma unroll 4
  for (int c = 0; c < C_PB; ++c) {
    const float* ar = A_lds + c * K_SEG;
    float4 v;
    v.x = ar[kk[0]] * mm[0];
    v.y = ar[kk[1]] * mm[1];
    v.z = ar[kk[2]] * mm[2];
    v.w = ar[kk[3]] * mm[3];
    if (full) {
      *(float4*)(obase + (long)c * T) = v;    // global_store_b128, coalesced
    } else {
      float tmp[4] = {v.x, v.y, v.z, v.w};
      for (int j = 0; j < 4; ++j)
        if (tb + j < (long)T) obase[(long)c * T + j] = tmp[j];
    }
  }
}

// ---------------------------------------------------------------------------
extern "C" void kernel_launch(void* const* d_in, const int* in_sizes, int n_in,
                              void* d_out, int out_size, void* d_ws, size_t ws_size,
                              hipStream_t stream) {
  const float* A = (const float*)d_in[0];   // [C, K] fp32
  const float* L = (const float*)d_in[1];   // [K]    fp32
  const int K = in_sizes[1];                // 256
  const int C = in_sizes[0] / K;            // 256
  const int T = out_size / C;               // 131072 (avoids device read of d_in[2])

  int*   cum  = (int*)d_ws;                 // 257 ints
  int*   meta = (int*)d_ws + 260;           // 2 ints
  float* Lp   = (float*)d_ws + 272;         // 256 floats

  plan_kernel<<<1, 256, 0, stream>>>(L, T, cum, meta, Lp);

  dim3 grid((unsigned)((T + T_PB - 1) / T_PB), 2u, 1u);
  sample_kernel<<<grid, 256, 0, stream>>>(A, cum, meta, Lp, (float*)d_out, T);
}